// GCNConv_78821239816696
// MI455X (gfx1250) — compile-verified
//
#include <hip/hip_runtime.h>

// ---------------------------------------------------------------------------
// GCNConv for MI455X (gfx1250, wave32).
//   h   = features @ W^T                  (fp32 GEMM -> V_WMMA_F32_16X16X4_F32)
//   pre = h*skip_weight + bias            (fused into GEMM epilogue)
//   pre += scatter_add(h[src]*w -> dst)   (fp32 global atomics, L2-resident h)
//   out = selu(pre)
// ---------------------------------------------------------------------------

typedef __attribute__((ext_vector_type(2))) float v2f;
typedef __attribute__((ext_vector_type(8))) float v8f;

#define N_NODES 50000
#define N_EDGES 800000
#define D 64  // D_IN == D_OUT == 64

// ---------------------------------------------------------------------------
// Kernel 1: one wave32 computes a 16x64 tile of h = feat @ W^T.
// A (16x4 f32 frag): lanes 0-15 hold M=lane, K=kb+{0,1}; lanes 16-31 hold
//                    M=lane-16, K=kb+{2,3}  -> contiguous float2 per lane.
// B (4x16 f32 frag): B[k][n] = W[n][k]; same half-split over K -> the lane's
//                    two elements are W[n0+l][kb+2*half .. +1], contiguous.
// C/D (16x16 f32):   VGPR r -> row m0 + r + 8*half, col n0 + (lane&15).
// ---------------------------------------------------------------------------
__global__ __launch_bounds__(256) void gemm_wmma(
    const float* __restrict__ feat,   // (N_NODES, D)
    const float* __restrict__ W,      // (D, D)  row-major (out, in)
    const float* __restrict__ skip_w, // (D,)
    const float* __restrict__ bias,   // (D,)
    float* __restrict__ h,            // (N_NODES, D) scratch: raw transform
    float* __restrict__ out_pre)      // (N_NODES, D) accumulator: skip + bias
{
  const int gwave = (int)((blockIdx.x * blockDim.x + threadIdx.x) >> 5);
  const int lane  = (int)(threadIdx.x & 31u);
  const int m0    = gwave * 16;
  if (m0 >= N_NODES) return;  // 50000/16 = 3125 waves exactly, guard anyway

  const int half = lane >> 4;   // 0: lanes 0-15, 1: lanes 16-31
  const int l    = lane & 15;

  v8f acc0 = {}, acc1 = {}, acc2 = {}, acc3 = {};

  const float* arow = feat + (size_t)(m0 + l) * D + 2 * half;
  const float* brow = W + (size_t)l * D + 2 * half;  // N-tile 0 base (n = l)

#pragma unroll
  for (int kb = 0; kb < D; kb += 4) {
    const v2f a = *(const v2f*)(arow + kb);
    const v2f b0 = *(const v2f*)(brow + kb);             // n0 = 0
    const v2f b1 = *(const v2f*)(brow + 16 * D + kb);    // n0 = 16
    const v2f b2 = *(const v2f*)(brow + 32 * D + kb);    // n0 = 32
    const v2f b3 = *(const v2f*)(brow + 48 * D + kb);    // n0 = 48
    acc0 = __builtin_amdgcn_wmma_f32_16x16x4_f32(false, a, false, b0,
                                                 (short)0, acc0, false, false);
    acc1 = __builtin_amdgcn_wmma_f32_16x16x4_f32(false, a, false, b1,
                                                 (short)0, acc1, false, false);
    acc2 = __builtin_amdgcn_wmma_f32_16x16x4_f32(false, a, false, b2,
                                                 (short)0, acc2, false, false);
    acc3 = __builtin_amdgcn_wmma_f32_16x16x4_f32(false, a, false, b3,
                                                 (short)0, acc3, false, false);
  }

  // Epilogue: write raw h (for the SpMM gather) and skip+bias (atomic target).
  const v8f accs[4] = {acc0, acc1, acc2, acc3};
#pragma unroll
  for (int nt = 0; nt < 4; ++nt) {
    const int n = nt * 16 + l;
    const float sw = skip_w[n];
    const float bs = bias[n];
#pragma unroll
    for (int r = 0; r < 8; ++r) {
      const int row = m0 + r + 8 * half;
      const size_t idx = (size_t)row * D + n;
      const float v = accs[nt][r];
      h[idx] = v;
      out_pre[idx] = v * sw + bs;
    }
  }
}

// ---------------------------------------------------------------------------
// Kernel 2: edge scatter. 16 lanes per edge; each lane gathers a float4 of
// h[src] (L2-resident: h is 12.8 MB << 192 MB L2) and issues 4 non-returning
// fp32 atomics into the destination row.
// ---------------------------------------------------------------------------
__global__ __launch_bounds__(256) void scatter_edges(
    const float* __restrict__ h,
    const float* __restrict__ ew,
    const int* __restrict__ esrc,
    const int* __restrict__ edst,
    float* __restrict__ out_pre)
{
  const long long t = (long long)blockIdx.x * blockDim.x + threadIdx.x;
  const long long e = t >> 4;
  const int q = (int)(t & 15);  // which float4 of the 64-float row
  if (e >= N_EDGES) return;

  const int src = esrc[e];
  const int dst = edst[e];
  const float w = ew[e];

  const float4 v = ((const float4*)(h + (size_t)src * D))[q];
  float* o = out_pre + (size_t)dst * D + q * 4;
  atomicAdd(o + 0, v.x * w);
  atomicAdd(o + 1, v.y * w);
  atomicAdd(o + 2, v.z * w);
  atomicAdd(o + 3, v.w * w);
}

// ---------------------------------------------------------------------------
// Kernel 3: SELU in place over the accumulated pre-activation.
// ---------------------------------------------------------------------------
__global__ __launch_bounds__(256) void selu_inplace(float* __restrict__ out,
                                                    int n)
{
  const int i = (int)(blockIdx.x * blockDim.x + threadIdx.x);
  if (i >= n) return;
  const float lam = 1.0507009873554805f;
  const float alp = 1.6732632423543772f;
  const float x = out[i];
  out[i] = lam * (x > 0.0f ? x : alp * (__expf(x) - 1.0f));
}

// ---------------------------------------------------------------------------
// Inputs (setup_inputs order):
//   0 features (50000*64 f32), 1 W (64*64 f32), 2 bias (64 f32),
//   3 skip_weight (64 f32), 4 edge_weight (800000 f32),
//   5 edge_src (800000 i32), 6 edge_dst (800000 i32)
// Output: (50000*64) f32. Workspace: h needs 12.8 MB.
// ---------------------------------------------------------------------------
extern "C" void kernel_launch(void* const* d_in, const int* in_sizes, int n_in,
                              void* d_out, int out_size, void* d_ws,
                              size_t ws_size, hipStream_t stream) {
  const float* feat   = (const float*)d_in[0];
  const float* W      = (const float*)d_in[1];
  const float* bias   = (const float*)d_in[2];
  const float* skip_w = (const float*)d_in[3];
  const float* ew     = (const float*)d_in[4];
  const int*   esrc   = (const int*)d_in[5];
  const int*   edst   = (const int*)d_in[6];
  float* out = (float*)d_out;
  float* h   = (float*)d_ws;  // N_NODES * D floats = 12.8 MB scratch

  // 3125 waves -> 100000 threads -> 391 blocks of 256 (8 waves/block).
  const int gemm_threads = ((N_NODES + 15) / 16) * 32;
  gemm_wmma<<<(gemm_threads + 255) / 256, 256, 0, stream>>>(feat, W, skip_w,
                                                            bias, h, out);

  const long long scat_threads = (long long)N_EDGES * 16;
  scatter_edges<<<(int)((scat_threads + 255) / 256), 256, 0, stream>>>(
      h, ew, esrc, edst, out);

  const int n = N_NODES * D;
  selu_inplace<<<(n + 255) / 256, 256, 0, stream>>>(out, n);
}